// GNNModel_24936580121235
// MI455X (gfx1250) — compile-verified
//
#include <hip/hip_runtime.h>
#include <math.h>

typedef __attribute__((ext_vector_type(2))) float v2f;
typedef __attribute__((ext_vector_type(8))) float v8f;

#define NN 100000
#define NE 3200000
#define INF_ 128
#define HID_ 64

// Guaranteed-hardware fp32 atomic add: non-returning global_atomic_add_f32,
// device scope (L2 atomic units), tracked with STOREcnt. Avoids any CAS-loop
// expansion of the generic atomic builtins.
__device__ __forceinline__ void dev_atomic_add_f32(float* p, float v) {
  asm volatile("global_atomic_add_f32 %0, %1, off scope:SCOPE_DEV"
               :
               : "v"(p), "v"(v)
               : "memory");
}

// ---------------- degree / normalization ----------------

__global__ __launch_bounds__(256) void k_set_one(float* __restrict__ deg, int n) {
  int i = blockIdx.x * blockDim.x + threadIdx.x;
  if (i < n) deg[i] = 1.0f;  // self-loop contribution
}

__global__ __launch_bounds__(256) void k_deg_accum(const int* __restrict__ dst,
                                                   float* __restrict__ deg, int e) {
  int i = blockIdx.x * blockDim.x + threadIdx.x;
  if (i < e) dev_atomic_add_f32(&deg[dst[i]], 1.0f);
}

__global__ __launch_bounds__(256) void k_rsqrt_inplace(float* __restrict__ d, int n) {
  int i = blockIdx.x * blockDim.x + threadIdx.x;
  if (i < n) d[i] = rsqrtf(d[i]);  // deg >= 1 always
}

// ---------------- WMMA fp32 GEMM: H[N,64] = X[N,K] @ W[K,64] ----------------
// One wave per 16-row tile; computes all four 16-col tiles of F=64.
// A layout (16x4 f32, ISA 7.12.2): lanes 0-15 hold K+0/K+1 of row M=lane,
// lanes 16-31 hold K+2/K+3 of row M=lane-16. B mirrored (rows across halves).
// C/D: VGPR v holds row half*8+v, col = lane%16.

__global__ __launch_bounds__(32) void k_gemm_wmma(const float* __restrict__ X,
                                                  const float* __restrict__ W,
                                                  float* __restrict__ H, int K) {
  const int F = 64;
  const int tileM = blockIdx.x * 16;
  const int lane = threadIdx.x;
  const int half = lane >> 4;
  const int l = lane & 15;

  v8f acc0 = {}, acc1 = {}, acc2 = {}, acc3 = {};
  const float* xrow = X + (tileM + l) * K + half * 2;

  for (int k0 = 0; k0 < K; k0 += 4) {
    v2f a;
    a.x = xrow[k0];
    a.y = xrow[k0 + 1];
    const float* wrow = W + (k0 + half * 2) * F + l;
    v2f b0; b0.x = wrow[0];      b0.y = wrow[F];
    v2f b1; b1.x = wrow[16];     b1.y = wrow[F + 16];
    v2f b2; b2.x = wrow[32];     b2.y = wrow[F + 32];
    v2f b3; b3.x = wrow[48];     b3.y = wrow[F + 48];
    acc0 = __builtin_amdgcn_wmma_f32_16x16x4_f32(false, a, false, b0, (short)0, acc0, false, false);
    acc1 = __builtin_amdgcn_wmma_f32_16x16x4_f32(false, a, false, b1, (short)0, acc1, false, false);
    acc2 = __builtin_amdgcn_wmma_f32_16x16x4_f32(false, a, false, b2, (short)0, acc2, false, false);
    acc3 = __builtin_amdgcn_wmma_f32_16x16x4_f32(false, a, false, b3, (short)0, acc3, false, false);
  }

  float* out = H + (tileM + half * 8) * F + l;
#pragma unroll
  for (int v = 0; v < 8; ++v) {
    out[v * F + 0]  = acc0[v];
    out[v * F + 16] = acc1[v];
    out[v * F + 32] = acc2[v];
    out[v * F + 48] = acc3[v];
  }
}

// ---------------- aggregation ----------------

// agg[i,:] = h[i,:] * dinv[i]^2   (self-loop term; also zero-initializes agg)
__global__ __launch_bounds__(256) void k_agg_init(const float* __restrict__ h,
                                                  const float* __restrict__ dinv,
                                                  float* __restrict__ agg) {
  int t = blockIdx.x * blockDim.x + threadIdx.x;  // NN*16 threads exactly
  int node = t >> 4;
  int f4 = (t & 15) * 4;
  float di = dinv[node];
  float w = di * di;
  const float4 hv = *reinterpret_cast<const float4*>(h + node * 64 + f4);
  float4 o;
  o.x = hv.x * w; o.y = hv.y * w; o.z = hv.z * w; o.w = hv.w * w;
  *reinterpret_cast<float4*>(agg + node * 64 + f4) = o;
}

// agg[dst,:] += h[src,:] * dinv[src]*dinv[dst]  (one thread per edge x 4 feats)
__global__ __launch_bounds__(256) void k_edge_scatter(const float* __restrict__ h,
                                                      const int* __restrict__ src,
                                                      const int* __restrict__ dst,
                                                      const float* __restrict__ dinv,
                                                      float* __restrict__ agg) {
  int t = blockIdx.x * blockDim.x + threadIdx.x;  // NE*16 threads exactly
  int e = t >> 4;
  int f4 = (t & 15) * 4;
  int s = src[e];
  int d = dst[e];
  float w = dinv[s] * dinv[d];
  const float4 hv = *reinterpret_cast<const float4*>(h + s * 64 + f4);
  float* out = agg + d * 64 + f4;
  dev_atomic_add_f32(out + 0, hv.x * w);
  dev_atomic_add_f32(out + 1, hv.y * w);
  dev_atomic_add_f32(out + 2, hv.z * w);
  dev_atomic_add_f32(out + 3, hv.w * w);
}

// out[i,:] = relu(in[i,:] + b[:])
__global__ __launch_bounds__(256) void k_bias_relu(const float* __restrict__ in,
                                                   const float* __restrict__ b,
                                                   float* __restrict__ out) {
  int t = blockIdx.x * blockDim.x + threadIdx.x;  // NN*16 threads exactly
  int node = t >> 4;
  int f4 = (t & 15) * 4;
  const float4 v = *reinterpret_cast<const float4*>(in + node * 64 + f4);
  float4 o;
  o.x = fmaxf(v.x + b[f4 + 0], 0.0f);
  o.y = fmaxf(v.y + b[f4 + 1], 0.0f);
  o.z = fmaxf(v.z + b[f4 + 2], 0.0f);
  o.w = fmaxf(v.w + b[f4 + 3], 0.0f);
  *reinterpret_cast<float4*>(out + node * 64 + f4) = o;
}

// out[i] = sigmoid(h[i,:] . wfc + bfc)   -- 16 lanes cooperate per node
__global__ __launch_bounds__(256) void k_fc_sigmoid(const float* __restrict__ h,
                                                    const float* __restrict__ wfc,
                                                    const float* __restrict__ bfc,
                                                    float* __restrict__ out) {
  int t = blockIdx.x * blockDim.x + threadIdx.x;  // NN*16 threads exactly
  int node = t >> 4;
  int g = t & 15;
  const float4 hv = *reinterpret_cast<const float4*>(h + node * 64 + g * 4);
  const float4 wv = *reinterpret_cast<const float4*>(wfc + g * 4);
  float p = hv.x * wv.x + hv.y * wv.y + hv.z * wv.z + hv.w * wv.w;
  p += __shfl_xor(p, 8, 16);
  p += __shfl_xor(p, 4, 16);
  p += __shfl_xor(p, 2, 16);
  p += __shfl_xor(p, 1, 16);
  if (g == 0) out[node] = 1.0f / (1.0f + __expf(-(p + bfc[0])));
}

// ---------------- launch ----------------

extern "C" void kernel_launch(void* const* d_in, const int* in_sizes, int n_in,
                              void* d_out, int out_size, void* d_ws, size_t ws_size,
                              hipStream_t stream) {
  const float* x   = (const float*)d_in[0];
  const int*   ei  = (const int*)d_in[1];
  const int*   src = ei;
  const int*   dst = ei + NE;
  const float* W1  = (const float*)d_in[2];
  const float* b1  = (const float*)d_in[3];
  const float* W2  = (const float*)d_in[4];
  const float* b2  = (const float*)d_in[5];
  const float* Wfc = (const float*)d_in[6];
  const float* bfc = (const float*)d_in[7];
  float* out = (float*)d_out;

  float* dinv = (float*)d_ws;
  float* bufA = dinv + ((NN + 255) & ~255);   // 100096-float offset, aligned
  float* bufB = bufA + (size_t)NN * 64;

  const int nodeBlk = (NN + 255) / 256;       // 391
  const int elemBlk = NN * 16 / 256;          // 6250 (exact)
  const int edgeBlk = NE / 256;               // 12500 (exact)
  const int scatBlk = (int)((long)NE * 16 / 256);  // 200000 (exact)

  // degree -> dinv
  k_set_one<<<nodeBlk, 256, 0, stream>>>(dinv, NN);
  k_deg_accum<<<edgeBlk, 256, 0, stream>>>(dst, dinv, NE);
  k_rsqrt_inplace<<<nodeBlk, 256, 0, stream>>>(dinv, NN);

  // layer 1: h1 = X@W1 ; agg = norm-propagate(h1) ; hr1 = relu(agg + b1)
  k_gemm_wmma<<<NN / 16, 32, 0, stream>>>(x, W1, bufA, INF_);
  k_agg_init<<<elemBlk, 256, 0, stream>>>(bufA, dinv, bufB);
  k_edge_scatter<<<scatBlk, 256, 0, stream>>>(bufA, src, dst, dinv, bufB);
  k_bias_relu<<<elemBlk, 256, 0, stream>>>(bufB, b1, bufA);

  // layer 2: h2 = hr1@W2 ; agg = norm-propagate(h2) ; hr2 = relu(agg + b2)
  k_gemm_wmma<<<NN / 16, 32, 0, stream>>>(bufA, W2, bufB, HID_);
  k_agg_init<<<elemBlk, 256, 0, stream>>>(bufB, dinv, bufA);
  k_edge_scatter<<<scatBlk, 256, 0, stream>>>(bufB, src, dst, dinv, bufA);
  k_bias_relu<<<elemBlk, 256, 0, stream>>>(bufA, b2, bufB);

  // fc + sigmoid
  k_fc_sigmoid<<<elemBlk, 256, 0, stream>>>(bufB, Wfc, bfc, out);
}